// Attn_head_89412629168239
// MI455X (gfx1250) — compile-verified
//
#include <hip/hip_runtime.h>

// Problem constants (from reference)
#define NNODE 55
#define KDIM  8192
#define ODIM  8192
// Tiling
#define KC     128            // K-chunk staged in LDS
#define LDSTR  132            // padded LDS row stride (floats): bank=(4r+c)%64 -> conflict-free
#define NBLK   32             // output channels per block (2 N-tiles of 16)
#define NCHUNK (KDIM / KC)    // 64

typedef float v2f __attribute__((ext_vector_type(2)));
typedef float v8f __attribute__((ext_vector_type(8)));
typedef unsigned int v4u __attribute__((ext_vector_type(4)));
typedef int v4i __attribute__((ext_vector_type(4)));
typedef int v8i __attribute__((ext_vector_type(8)));

#if defined(__HIP_DEVICE_COMPILE__) && __has_builtin(__builtin_amdgcn_tensor_load_to_lds)
#define USE_TDM 1
#else
#define USE_TDM 0
#endif

#if USE_TDM
// Issue one 2D TDM tile load (global -> LDS) per CDNA5 ISA ch.8 D# layout.
// data_size=4B; LDS padding: every 128 dwords insert 4 dwords -> row stride 132.
__device__ __forceinline__ void tdm_load_tile(const float* gaddr, unsigned lds_off,
                                              unsigned td0, unsigned td1,
                                              unsigned tile0, unsigned tile1,
                                              unsigned long long stride0_elems) {
  unsigned long long ga = (unsigned long long)(uintptr_t)gaddr;
  v4u g0;
  g0.x = 1u;                                                   // count=1 (valid descriptor)
  g0.y = lds_off;                                              // LDS byte offset
  g0.z = (unsigned)(ga & 0xFFFFFFFFull);                       // global_addr[31:0]
  g0.w = (unsigned)((ga >> 32) & 0x01FFFFFFull) | (2u << 30);  // global_addr[56:32] | type=2
  v8i g1;
  // data_size=2(4B)<<16 | pad_enable<<20 | pad_interval=6(128dw)<<22 | pad_amount=3(4dw)<<25
  g1[0] = (int)((2u << 16) | (1u << 20) | (6u << 22) | (3u << 25));
  g1[1] = (int)((td0 & 0xFFFFu) << 16);                        // tensor_dim0[15:0]
  g1[2] = (int)((td0 >> 16) | ((td1 & 0xFFFFu) << 16));        // td0[31:16] | td1[15:0]
  g1[3] = (int)((td1 >> 16) | (tile0 << 16));                  // td1[31:16] | tile_dim0
  g1[4] = (int)(tile1 & 0xFFFFu);                              // tile_dim1 (tile_dim2=0 -> 2D)
  g1[5] = (int)(unsigned)(stride0_elems & 0xFFFFFFFFull);      // tensor_dim0_stride[31:0]
  g1[6] = (int)(unsigned)((stride0_elems >> 32) & 0xFFFFull);  // stride0[47:32] (stride1=0)
  g1[7] = 0;
  v4i z4 = {0, 0, 0, 0};
#if defined(__clang_major__) && (__clang_major__ >= 23)
  v8i z8 = {0, 0, 0, 0, 0, 0, 0, 0};
  __builtin_amdgcn_tensor_load_to_lds(g0, g1, z4, z4, z8, 0);
#else
  __builtin_amdgcn_tensor_load_to_lds(g0, g1, z4, z4, 0);
#endif
}
#endif  // USE_TDM

// out[m, o] = elu( sum_k x[m,k] * W1[o,k] + b1[o] ), m < 55
__global__ __launch_bounds__(256, 2)
void attn_head_gemm_elu(const float* __restrict__ x, const float* __restrict__ W1,
                        const float* __restrict__ b1, float* __restrict__ out) {
#if defined(__HIP_DEVICE_COMPILE__)
  __shared__ __align__(16) float Alds[2][64 * LDSTR];    // x chunk:  64 rows x 128 k (padded)
  __shared__ __align__(16) float Blds[2][NBLK * LDSTR];  // W1 chunk: 32 rows x 128 k (padded)

  const int tid  = threadIdx.x;
  const int lane = tid & 31;
  const int wv   = tid >> 5;
  const int lr   = lane & 15;   // row/col within tile
  const int hl   = lane >> 4;   // lane half selects K pair (A/B) / M+8 (C)
  const int mt   = wv >> 1;     // 0..3  -> node rows mt*16..mt*16+15
  const int nt   = wv & 1;      // 0..1  -> out cols nt*16..nt*16+15 within slab
  const int obase = blockIdx.x * NBLK;

  v8f acc = {0.f, 0.f, 0.f, 0.f, 0.f, 0.f, 0.f, 0.f};

#if USE_TDM
  if (tid < 32) {  // wave 0 drives the Tensor Data Mover
    tdm_load_tile(x, (unsigned)(uintptr_t)&Alds[0][0], KDIM, NNODE, KC, 64, KDIM);
    tdm_load_tile(W1 + (size_t)obase * KDIM, (unsigned)(uintptr_t)&Blds[0][0],
                  KDIM, (unsigned)(ODIM - obase), KC, NBLK, KDIM);
    __builtin_amdgcn_s_wait_tensorcnt(0);
  }
  __syncthreads();
#endif

  for (int c = 0; c < NCHUNK; ++c) {
    const int cur = c & 1;
#if USE_TDM
    if (c + 1 < NCHUNK && tid < 32) {  // prefetch next chunk into other buffer
      const int nk = (c + 1) * KC;
      tdm_load_tile(x + nk, (unsigned)(uintptr_t)&Alds[cur ^ 1][0],
                    (unsigned)(KDIM - nk), NNODE, KC, 64, KDIM);
      tdm_load_tile(W1 + (size_t)obase * KDIM + nk, (unsigned)(uintptr_t)&Blds[cur ^ 1][0],
                    (unsigned)(KDIM - nk), (unsigned)(ODIM - obase), KC, NBLK, KDIM);
    }
#else
    {  // fallback: cooperative coalesced staging (float4), zero-pad rows >= 55
      const int k0 = c * KC;
      for (int idx = tid * 4; idx < 64 * KC; idx += 256 * 4) {
        const int r = idx / KC, cc = idx % KC;
        float4 vv = make_float4(0.f, 0.f, 0.f, 0.f);
        if (r < NNODE) vv = *(const float4*)(x + (size_t)r * KDIM + k0 + cc);
        *(float4*)(&Alds[cur][r * LDSTR + cc]) = vv;
      }
      for (int idx = tid * 4; idx < NBLK * KC; idx += 256 * 4) {
        const int r = idx / KC, cc = idx % KC;
        *(float4*)(&Blds[cur][r * LDSTR + cc]) =
            *(const float4*)(W1 + (size_t)(obase + r) * KDIM + k0 + cc);
      }
      __syncthreads();
    }
#endif

    // A frag (16x4 f32): lane l<16 -> row l, k={0,1}; lane l>=16 -> row l-16, k={2,3}
    // B frag (4x16):     lane l<16 -> col l, k={0,1}; lane l>=16 -> col l-16, k={2,3}
    // Both reduce to: float2 at [tile*16 + lr][ks + 2*hl] in the padded LDS tiles.
    const float* Ab = &Alds[cur][(mt * 16 + lr) * LDSTR + 2 * hl];
    const float* Bb = &Blds[cur][(nt * 16 + lr) * LDSTR + 2 * hl];
#pragma unroll
    for (int ks = 0; ks < KC; ks += 4) {
      v2f af = *(const v2f*)(Ab + ks);
      v2f bf = *(const v2f*)(Bb + ks);
      acc = __builtin_amdgcn_wmma_f32_16x16x4_f32(false, af, false, bf,
                                                  (short)0, acc, false, false);
    }

#if USE_TDM
    if (c + 1 < NCHUNK && tid < 32) __builtin_amdgcn_s_wait_tensorcnt(0);
#endif
    __syncthreads();
  }

  // C layout: VGPR v, lanes 0-15 -> M=v, N=lane; lanes 16-31 -> M=v+8, N=lane-16
  const int o = obase + nt * 16 + lr;
  const float bias = b1[o];
#pragma unroll
  for (int vi = 0; vi < 8; ++vi) {
    const int m = mt * 16 + hl * 8 + vi;
    if (m < NNODE) {
      const float r = acc[vi] + bias;
      out[(size_t)m * ODIM + o] = (r > 0.f) ? r : (__expf(r) - 1.0f);  // ELU(alpha=1)
    }
  }
#endif  // __HIP_DEVICE_COMPILE__
}

extern "C" void kernel_launch(void* const* d_in, const int* in_sizes, int n_in,
                              void* d_out, int out_size, void* d_ws, size_t ws_size,
                              hipStream_t stream) {
  (void)in_sizes; (void)n_in; (void)out_size; (void)d_ws; (void)ws_size;
  // setup_inputs order: x, adj, p_covs, W1, b1, a1, ba1, a2, ba2
  // adj/p_covs/a1/ba1/a2/ba2 are dead: softmax over a size-1 axis == 1.
  const float* x  = (const float*)d_in[0];
  const float* W1 = (const float*)d_in[3];
  const float* b1 = (const float*)d_in[4];
  float* out = (float*)d_out;
  dim3 grid(ODIM / NBLK);  // 256 blocks
  dim3 block(256);         // 8 wave32 waves
  attn_head_gemm_elu<<<grid, block, 0, stream>>>(x, W1, b1, out);
}